// mean_aggregator_49795850830175
// MI455X (gfx1250) — compile-verified
//
#include <hip/hip_runtime.h>
#include <stdint.h>

// Mean aggregator: out[b, :] = mean_s emb[neighbors[b,s], :]
// B=50000 rows, S=32 samples, D=128 dims, f32.
// Memory-bound gather (~850 MB @ 23.3 TB/s ~= 37us floor). Strategy:
// per-row wave32, async-gather 32 embedding rows (512B each, fully
// coalesced: 32 lanes x b128) into LDS via ASYNCcnt-tracked
// global_load_async_to_lds_b128, then reduce from LDS with float4 adds.
// Index reads and output writes are non-temporal so the 256MB embedding
// table keeps maximal residency in the 192MB L2 (avg ~3.2x reuse/node).

#define S_FANOUT 32
#define D_DIM    128
#define WAVES    2
#define TPB      (WAVES * 32)

typedef int   v4i __attribute__((ext_vector_type(4)));
typedef float v4f __attribute__((ext_vector_type(4)));
typedef __attribute__((address_space(1))) v4i GV4;  // global int4
typedef __attribute__((address_space(3))) v4i LV4;  // LDS int4
typedef __attribute__((address_space(3))) void LASV;

__device__ __forceinline__ void async_gather_b128(const void* gsrc, void* ldst) {
#if defined(__AMDGCN__) && __has_builtin(__builtin_amdgcn_global_load_async_to_lds_b128)
  __builtin_amdgcn_global_load_async_to_lds_b128((GV4*)gsrc, (LV4*)ldst,
                                                 /*offset=*/0, /*cpol=*/0);
#else
  unsigned lds_addr = (unsigned)(unsigned long long)(LASV*)ldst;
  asm volatile("global_load_async_to_lds_b128 %0, %1, off"
               :
               : "v"(lds_addr), "v"((unsigned long long)gsrc)
               : "memory");
#endif
}

__device__ __forceinline__ void wait_async_zero() {
#if defined(__AMDGCN__) && __has_builtin(__builtin_amdgcn_s_wait_asynccnt)
  __builtin_amdgcn_s_wait_asynccnt(0);
#else
  asm volatile("s_wait_asynccnt 0x0" ::: "memory");
#endif
}

__global__ __launch_bounds__(TPB) void mean_aggregator_49795850830175_kernel(
    const int* __restrict__ neighbors,   // [n_rows, 32] int32
    const float* __restrict__ emb,       // [500000, 128] f32
    float* __restrict__ out,             // [n_rows, 128] f32
    int n_rows) {
  __shared__ float lds[WAVES * S_FANOUT * D_DIM];  // 2 * 32 * 128 * 4B = 32 KB

  const int lane = threadIdx.x & 31;
  const int wave = threadIdx.x >> 5;
  const int row  = blockIdx.x * WAVES + wave;
  if (row >= n_rows) return;

  float* buf = &lds[wave * (S_FANOUT * D_DIM)];

  // Coalesced, read-once -> non-temporal: lane l holds neighbor s=l.
  const int nidx = __builtin_nontemporal_load(&neighbors[row * S_FANOUT + lane]);

  // Issue 32 async row-gathers. Each instruction moves one full 512B
  // embedding row: lane l carries bytes [l*16, l*16+16). Default (RT)
  // temporal hint keeps embedding lines resident in L2 for reuse.
#pragma unroll
  for (int s = 0; s < S_FANOUT; ++s) {
    const unsigned ns = (unsigned)__builtin_amdgcn_readlane(nidx, s);
    const float* src = emb + (unsigned long long)ns * D_DIM + lane * 4;
    async_gather_b128(src, buf + s * D_DIM + lane * 4);
  }

  wait_async_zero();

  // Reduce 32 rows from LDS; lane l owns dims [4l, 4l+4).
  v4f acc = {0.f, 0.f, 0.f, 0.f};
#pragma unroll
  for (int s = 0; s < S_FANOUT; ++s) {
    const v4f v = *(const v4f*)(buf + s * D_DIM + lane * 4);
    acc += v;
  }
  const float inv = 1.0f / (float)S_FANOUT;
  v4f r = acc * inv;

  // Write-once result stream -> non-temporal store (don't evict emb from L2).
  v4f* dst = (v4f*)(out + (unsigned long long)row * D_DIM + lane * 4);
  __builtin_nontemporal_store(r, dst);
}

extern "C" void kernel_launch(void* const* d_in, const int* in_sizes, int n_in,
                              void* d_out, int out_size, void* d_ws, size_t ws_size,
                              hipStream_t stream) {
  (void)n_in; (void)out_size; (void)d_ws; (void)ws_size;
  const int* neighbors = (const int*)d_in[0];
  const float* emb     = (const float*)d_in[1];
  float* out           = (float*)d_out;

  const int n_rows = in_sizes[0] / S_FANOUT;  // 50000
  const int blocks = (n_rows + WAVES - 1) / WAVES;
  hipLaunchKernelGGL(mean_aggregator_49795850830175_kernel,
                     dim3(blocks), dim3(TPB), 0, stream,
                     neighbors, emb, out, n_rows);
}